// BallQueryHO_19774029431169
// MI455X (gfx1250) — compile-verified
//
#include <hip/hip_runtime.h>

typedef float v2f __attribute__((ext_vector_type(2)));
typedef float v8f __attribute__((ext_vector_type(8)));

#define BQ_B 4
#define BQ_N 16384
#define BQ_M 4096
#define BQ_K 32
#define BQ_R2 0.01f

// One wave handles a tile of 16 centers; d2 for 16 centers x 16 points per
// V_WMMA_F32_16X16X4_F32. A = [-2cx,-2cy,-2cz,|c|^2], B = [px,py,pz,1],
// C preloaded with |p|^2 broadcast per column, so D = squared distances.
__global__ __launch_bounds__(128) void ball_query_wmma_kernel(
    const float* __restrict__ points,    // (B,3,N)
    const float* __restrict__ centers,   // (B,3,M)
    float* __restrict__ out)             // (B, 6*K, M)
{
    const int lane = threadIdx.x & 31;
    const int wave = threadIdx.x >> 5;
    const int tile = blockIdx.x * 4 + wave;          // 0 .. B*M/16 - 1
    const int b    = tile / (BQ_M / 16);
    const int m0   = (tile % (BQ_M / 16)) * 16;

    const float* px = points + (size_t)b * 3 * BQ_N;
    const float* py = px + BQ_N;
    const float* pz = py + BQ_N;
    const float* cx = centers + (size_t)b * 3 * BQ_M;
    const float* cy = cx + BQ_M;
    const float* cz = cy + BQ_M;

    // Each lane owns center m0 + (lane&15): lanes 0-15 carry A K0/K1,
    // lanes 16-31 carry A K2/K3 for the same 16 rows (ISA 16x4 A layout).
    const int   mloc = lane & 15;
    const int   m    = m0 + mloc;
    const float ccx = cx[m], ccy = cy[m], ccz = cz[m];
    const float c2  = ccx * ccx + ccy * ccy + ccz * ccz;

    v2f amat;
    if (lane < 16) { amat.x = -2.0f * ccx; amat.y = -2.0f * ccy; }
    else           { amat.x = -2.0f * ccz; amat.y = c2;          }

    // Lane r (<16) performs first-K selection for row r; lanes 16-31 idle
    // during selection and report "full" so the early-exit ballot works.
    unsigned cnt = (lane < 16) ? 0u : (unsigned)BQ_K;
    float* outb = out + (size_t)b * 6 * BQ_K * BQ_M + m;

    bool early = false;
    for (int n0 = 0; n0 < BQ_N; n0 += 16) {
        const int n = n0 + (lane & 15);
        const float qx = px[n], qy = py[n], qz = pz[n];
        const float p2 = qx * qx + qy * qy + qz * qz;

        // WGP-scope prefetch of upcoming point tiles (locality 3 -> pull
        // into all cache levels, not just GL2).
        __builtin_prefetch(px + n0 + 512, 0, 3);
        __builtin_prefetch(py + n0 + 512, 0, 3);
        __builtin_prefetch(pz + n0 + 512, 0, 3);

        v2f bmat;
        if (lane < 16) { bmat.x = qx; bmat.y = qy;   }
        else           { bmat.x = qz; bmat.y = 1.0f; }

        v8f cacc = { p2, p2, p2, p2, p2, p2, p2, p2 };

        // D[m][n] = -2 c.p + |c|^2 + |p|^2  (full f32, 256 pair evals)
        v8f d = __builtin_amdgcn_wmma_f32_16x16x4_f32(
            false, amat, false, bmat, (short)0, cacc, false, false);

        // Fast path: for tiles with zero in-ball pairs, one 8-way
        // fmin + ballot skips the whole mask-assembly block for them.
        float dmin = fminf(fminf(fminf(d[0], d[1]), fminf(d[2], d[3])),
                           fminf(fminf(d[4], d[5]), fminf(d[6], d[7])));
        if (__builtin_amdgcn_ballot_w32(dmin < BQ_R2) != 0u) {
            // Convert the 16x16 distance tile into per-row 16-bit masks:
            // ballot bits 0-15 = row v, bits 16-31 = row v+8 (C/D layout).
            unsigned mymask = 0u;
#pragma unroll
            for (int v = 0; v < 8; ++v) {
                const unsigned bal = __builtin_amdgcn_ballot_w32(d[v] < BQ_R2);
                if ((lane & 7) == v)
                    mymask = (lane < 8) ? (bal & 0xFFFFu) : (bal >> 16);
            }
            if (lane >= 16) mymask = 0u;

            // Append hits in ascending point-index order until K filled.
            while (mymask && cnt < BQ_K) {
                const int bit = __builtin_ctz(mymask);
                mymask &= mymask - 1u;
                const int nn = n0 + bit;
                const float sx = px[nn], sy = py[nn], sz = pz[nn];
                outb[(0 * BQ_K + cnt) * BQ_M] = sx;
                outb[(1 * BQ_K + cnt) * BQ_M] = sy;
                outb[(2 * BQ_K + cnt) * BQ_M] = sz;
                outb[(3 * BQ_K + cnt) * BQ_M] = sx - ccx;
                outb[(4 * BQ_K + cnt) * BQ_M] = sy - ccy;
                outb[(5 * BQ_K + cnt) * BQ_M] = sz - ccz;
                ++cnt;
            }

            // Uniform early exit once every row has K neighbors.
            if (__builtin_amdgcn_ballot_w32(cnt >= BQ_K) == 0xFFFFFFFFu) {
                early = true;
                break;
            }
        }
    }

    // Pad unfilled slots with point index 0 (matches the JAX zero-scatter).
    if (!early && lane < 16 && cnt < BQ_K) {
        const float sx = px[0], sy = py[0], sz = pz[0];
        for (unsigned j = cnt; j < (unsigned)BQ_K; ++j) {
            outb[(0 * BQ_K + j) * BQ_M] = sx;
            outb[(1 * BQ_K + j) * BQ_M] = sy;
            outb[(2 * BQ_K + j) * BQ_M] = sz;
            outb[(3 * BQ_K + j) * BQ_M] = sx - ccx;
            outb[(4 * BQ_K + j) * BQ_M] = sy - ccy;
            outb[(5 * BQ_K + j) * BQ_M] = sz - ccz;
        }
    }
}

extern "C" void kernel_launch(void* const* d_in, const int* in_sizes, int n_in,
                              void* d_out, int out_size, void* d_ws, size_t ws_size,
                              hipStream_t stream) {
    (void)in_sizes; (void)n_in; (void)out_size; (void)d_ws; (void)ws_size;
    const float* points  = (const float*)d_in[0];   // (4,3,16384) f32
    const float* centers = (const float*)d_in[1];   // (4,3,4096)  f32
    float* out = (float*)d_out;                     // (4,192,4096) f32

    const int tiles  = BQ_B * (BQ_M / 16);          // 1024 waves of work
    const int blocks = tiles / 4;                   // 4 waves (128 thr) per block
    ball_query_wmma_kernel<<<blocks, 128, 0, stream>>>(points, centers, out);
}